// SingleGAP_35038343200902
// MI455X (gfx1250) — compile-verified
//
#include <hip/hip_runtime.h>
#include <math.h>

#define KB2 240   // K2 block count (fixed -> deterministic reduction shape)

typedef __attribute__((ext_vector_type(2))) float v2f;
typedef __attribute__((ext_vector_type(8))) float v8f;

// Order-preserving float->uint map (for integer atomicMax on floats)
__device__ __forceinline__ unsigned f32_ord(float f) {
    unsigned b = __float_as_uint(f);
    return (b & 0x80000000u) ? ~b : (b | 0x80000000u);
}
__device__ __forceinline__ float ord_f32(unsigned u) {
    return __uint_as_float((u & 0x80000000u) ? (u ^ 0x80000000u) : ~u);
}

// ---------------------------------------------------------------------------
// K0: zero max/denom header AND precompute the folded center-MLP constants:
//   weff[d] = sum_f W1[f]*Wf[f,d]   -> ws_f[40..55]
//   ceff    = W1.bf + b1            -> ws_f[56]
// so the 16K K1 waves don't each redo 256 MACs of uniform setup.
// ---------------------------------------------------------------------------
__global__ void k0_init(unsigned* __restrict__ ws,
                        const float* __restrict__ Wf, const float* __restrict__ bf,
                        const float* __restrict__ W1, const float* __restrict__ b1p) {
    const int t = threadIdx.x;
    if (t < 32) ws[t] = 0u;
    float* wsf = (float*)ws;
    if (t < 16) {
        float s = 0.f;
        for (int f = 0; f < 16; ++f) s += W1[f] * Wf[f * 16 + t];
        wsf[40 + t] = s;
    }
    if (t == 0) {
        float c = b1p[0];
        for (int f = 0; f < 16; ++f) c += W1[f] * bf[f];
        wsf[56] = c;
    }
}

// ---------------------------------------------------------------------------
// K1: fused main pass over FULL 16-point groups (no predication, no tail).
// Wave handles 16 points: D(16x16) = Wf(16x16) x Xcols(16x16) + bf via
// 4 chained V_WMMA_F32_16X16X4_F32 per 16-column tile, 10 tiles per group.
// All addressing = uniform SGPR base + 32-bit lane offset (SADDR form).
// B fragments for a whole tile are loaded BEFORE the WMMA phase so loads
// batch into clauses and waits become incremental.
// x2v / xknn / x are streaming -> non-temporal; logits stay RT (L2-resident).
// ---------------------------------------------------------------------------
__global__ void k1_main(const float* __restrict__ xknn, const float* __restrict__ x,
                        const float* __restrict__ Wf,  const float* __restrict__ bf,
                        const float* __restrict__ W1,  const float* __restrict__ b1p,
                        const float* __restrict__ cst,  // ws floats: weff@40, ceff@56
                        float* __restrict__ out_logits, float* __restrict__ out_x2v,
                        unsigned* __restrict__ ws_max, int nFullGroups)
{
    const int lane   = threadIdx.x & 31;
    const int lane15 = lane & 15;
    const int hi     = lane >> 4;                 // 0: rows 0-7 / K 0-1 ; 1: rows 8-15 / K 2-3
    const int wavesPerBlock = blockDim.x >> 5;
    const int wave      = blockIdx.x * wavesPerBlock + (threadIdx.x >> 5);
    const int waveCount = gridDim.x * wavesPerBlock;

    const float b1   = b1p[0];
    const float ceff = cst[56];

    // A fragments (Wf), hoisted
    v2f afrag[4];
    {
        const int m = lane15;
        #pragma unroll
        for (int q = 0; q < 4; ++q) {
            const int d0 = 4 * q + 2 * hi;
            afrag[q].x = Wf[m * 16 + d0];
            afrag[q].y = Wf[m * 16 + d0 + 1];
        }
    }
    // C init fragment (bias bf broadcast along columns)
    v8f cfrag;
    {
        const int fb = hi * 8;
        #pragma unroll
        for (int r = 0; r < 8; ++r) cfrag[r] = bf[fb + r];
    }
    // W1 slice matching this half's row set (for a2 = W1 . h2)
    float w1s[8];
    #pragma unroll
    for (int r = 0; r < 8; ++r) w1s[r] = W1[hi * 8 + r];

    const int p0 = lane15 / 10;
    const int k0 = lane15 % 10;

    float wmax = -3.0e38f;

    for (int g = wave; g < nFullGroups; g += waveCount) {
        const int n0 = g * 16;
        const float* gb = xknn       + (size_t)n0 * 160;   // uniform bases (SADDR)
        float*       ob = out_x2v    + (size_t)n0 * 160;
        float*       lb = out_logits + (size_t)n0 * 16;

        // a1 for point p = lane15 (weff uniform -> SGPR MACs; x streamed NT)
        float a1v;
        {
            const float* xp = x + (size_t)(n0 + lane15) * 16;
            float s = ceff;
            #pragma unroll
            for (int d = 0; d < 16; ++d)
                s += cst[40 + d] * __builtin_nontemporal_load(xp + d);
            a1v = s;
        }

        // Column walk: global column gc = 16*c + lane15 ; point p = gc/10, k = gc%10
        int p = p0, k = k0;
        #pragma unroll 2
        for (int c = 0; c < 10; ++c) {
            const unsigned off = (unsigned)(p * 160 + hi * 20 + k);
            // Load ALL B fragments of this tile first (clause-batched)
            v2f bfr[4];
            #pragma unroll
            for (int q = 0; q < 4; ++q) {
                bfr[q].x = __builtin_nontemporal_load(gb + off + q * 40);       // d = 4q+2hi
                bfr[q].y = __builtin_nontemporal_load(gb + off + q * 40 + 10);  // d = 4q+2hi+1
            }
            // WMMA phase: K=16 contraction as 4 chained K=4 steps
            v8f acc = cfrag;
            #pragma unroll
            for (int q = 0; q < 4; ++q)
                acc = __builtin_amdgcn_wmma_f32_16x16x4_f32(
                        false, afrag[q], false, bfr[q], (short)0, acc, false, false);

            // x2v raw-reshape store: out2[n*160 + f*10 + k], f = hi*8 + r (streaming NT)
            const unsigned so = (unsigned)(p * 160 + hi * 80 + k);
            float s = 0.f;
            #pragma unroll
            for (int r = 0; r < 8; ++r) {
                __builtin_nontemporal_store(acc[r], ob + so + r * 10);
                s += w1s[r] * acc[r];
            }
            // a2 needs both row-halves of the column: cross-half add (wave32)
            s += __shfl_xor(s, 16, 32);
            // a1 of this column's point lives in lane p
            const float a1p = __shfl(a1v, p, 32);
            const float logit = s + b1 + a1p;
            if (hi == 0) lb[(unsigned)(p * 16 + k)] = logit;  // RT: stays in 192MB L2
            wmax = fmaxf(wmax, logit);

            // advance 16 columns
            p += 1; k += 6; if (k >= 10) { k -= 10; p += 1; }
        }
    }

    // Wave max reduce, then deterministic integer atomicMax
    #pragma unroll
    for (int off = 16; off >= 1; off >>= 1)
        wmax = fmaxf(wmax, __shfl_xor(wmax, off, 32));
    if (lane == 0) atomicMax(ws_max, f32_ord(wmax));
}

// ---------------------------------------------------------------------------
// K1t: scalar tail for the (rare) N % 16 leftover points. Cold path.
// ---------------------------------------------------------------------------
__global__ void k1_tail(const float* __restrict__ xknn, const float* __restrict__ x,
                        const float* __restrict__ Wf,  const float* __restrict__ bf,
                        const float* __restrict__ W1,  const float* __restrict__ b1p,
                        float* __restrict__ out_logits, float* __restrict__ out_x2v,
                        unsigned* __restrict__ ws_max, int nStart, int N)
{
    const int n = nStart + (int)(blockIdx.x * blockDim.x + threadIdx.x);
    if (n < N) {
        float lmax = -3.0e38f;
        const float b1 = b1p[0];
        float a1 = b1;
        for (int f = 0; f < 16; ++f) {
            float s = bf[f];
            for (int d = 0; d < 16; ++d) s += Wf[f * 16 + d] * x[(size_t)n * 16 + d];
            a1 += W1[f] * s;
        }
        for (int k = 0; k < 10; ++k) {
            float a2 = b1;
            for (int f = 0; f < 16; ++f) {
                float s = bf[f];
                for (int d = 0; d < 16; ++d)
                    s += Wf[f * 16 + d] * xknn[(size_t)n * 160 + d * 10 + k];
                out_x2v[(size_t)n * 160 + f * 10 + k] = s;
                a2 += W1[f] * s;
            }
            const float logit = a2 + a1;
            out_logits[(size_t)n * 16 + k] = logit;
            lmax = fmaxf(lmax, logit);
        }
        atomicMax(ws_max, f32_ord(lmax));
    }
}

// ---------------------------------------------------------------------------
// K2: per-k partial sums of exp(logit - M); fixed KB2 blocks, deterministic.
// Logits should still be L2-resident (regular TH, 40MB << 192MB L2).
// ---------------------------------------------------------------------------
__global__ void k2_partial(const float* __restrict__ out_logits,
                           const unsigned* __restrict__ ws_max,
                           float* __restrict__ partial, int N)
{
    const float M = ord_f32(*ws_max);
    float s[10];
    #pragma unroll
    for (int k = 0; k < 10; ++k) s[k] = 0.f;

    const int stride = gridDim.x * blockDim.x;
    for (int n = blockIdx.x * blockDim.x + threadIdx.x; n < N; n += stride) {
        const float* lp = out_logits + (size_t)n * 16;
        #pragma unroll
        for (int k = 0; k < 10; ++k) s[k] += __expf(lp[k] - M);
    }

    __shared__ float red[256];
    for (int k = 0; k < 10; ++k) {
        red[threadIdx.x] = s[k];
        __syncthreads();
        for (int off = 128; off >= 1; off >>= 1) {
            if ((int)threadIdx.x < off) red[threadIdx.x] += red[threadIdx.x + off];
            __syncthreads();
        }
        if (threadIdx.x == 0) partial[blockIdx.x * 10 + k] = red[0];
        __syncthreads();
    }
}

// K2b: fold KB2 partials per k, fixed order.
__global__ void k2b_fold(const float* __restrict__ partial, float* __restrict__ denom)
{
    const int k = threadIdx.x;
    if (k < 10) {
        float s = 0.f;
        for (int b = 0; b < KB2; ++b) s += partial[b * 10 + k];
        denom[k] = s;
    }
}

// ---------------------------------------------------------------------------
// K3: half-wave per point. Reads logits from out1 slots (L2-resident),
// x2v coalesced non-temporal, overwrites out1 with x_attn.
// Same-wave lockstep makes the read-then-overwrite of out1 safe.
// ---------------------------------------------------------------------------
__global__ void k3_final(float* __restrict__ out1, const float* __restrict__ x2v,
                         const unsigned* __restrict__ ws_max,
                         const float* __restrict__ denom, int N)
{
    const float M = ord_f32(*ws_max);
    float rdk[10];
    #pragma unroll
    for (int k = 0; k < 10; ++k) rdk[k] = 1.0f / denom[k];

    const int stride = gridDim.x * blockDim.x;
    const int total  = N * 16;
    for (int t = blockIdx.x * blockDim.x + threadIdx.x; t < total; t += stride) {
        const int n = t >> 4;
        const int j = t & 15;
        const float* lp = out1 + (size_t)n * 16;
        float xc[10];
        #pragma unroll
        for (int k = 0; k < 10; ++k) xc[k] = __expf(lp[k] - M) * rdk[k];
        const float* vp = x2v + (size_t)n * 160 + j;
        float acc = 0.f;
        #pragma unroll
        for (int k = 0; k < 10; ++k) acc += xc[k] * __builtin_nontemporal_load(vp + k * 16);
        __builtin_nontemporal_store(acc, out1 + (size_t)n * 16 + j);
    }
}

// ---------------------------------------------------------------------------
extern "C" void kernel_launch(void* const* d_in, const int* in_sizes, int n_in,
                              void* d_out, int out_size, void* d_ws, size_t ws_size,
                              hipStream_t stream) {
    const float* xknn = (const float*)d_in[0];   // [N,16,10]
    const float* x    = (const float*)d_in[1];   // [N,16]
    const float* Wf   = (const float*)d_in[2];   // [16,16]
    const float* bf   = (const float*)d_in[3];   // [16]
    const float* W1   = (const float*)d_in[4];   // [16]
    const float* b1   = (const float*)d_in[5];   // scalar

    const int N = in_sizes[1] / 16;

    float* out1 = (float*)d_out;                 // x_attn [N,16] (logit scratch first)
    float* out2 = out1 + (size_t)N * 16;         // x2v    [N,10,16]

    unsigned* ws_max   = (unsigned*)d_ws;        // [0]
    float*    ws_denom = (float*)d_ws + 16;      // [16..25]
    float*    ws_part  = (float*)d_ws + 32;      // [32 .. 32+KB2*10)
    const float* cst   = (const float*)d_ws;     // weff@40, ceff@56

    const int nFullGroups = N / 16;              // full 16-point groups (no predication)
    const int tailStart   = nFullGroups * 16;

    k0_init<<<1, 64, 0, stream>>>(ws_max, Wf, bf, W1, b1);
    if (nFullGroups > 0)
        k1_main<<<2048, 256, 0, stream>>>(xknn, x, Wf, bf, W1, b1, cst,
                                          out1, out2, ws_max, nFullGroups);
    if (tailStart < N)
        k1_tail<<<1, 32, 0, stream>>>(xknn, x, Wf, bf, W1, b1,
                                      out1, out2, ws_max, tailStart, N);
    k2_partial<<<KB2, 256, 0, stream>>>(out1, ws_max, ws_part, N);
    k2b_fold<<<1, 64, 0, stream>>>(ws_part, ws_denom);
    k3_final<<<2048, 256, 0, stream>>>(out1, out2, ws_max, ws_denom, N);
}